// MCPBRNN_Generic_PETconstraint_Scaling_10196252360838
// MI455X (gfx1250) — compile-verified
//
#include <hip/hip_runtime.h>
#include <math.h>

#define SPIN_LEN 365
#define TRAIN_LEN 4000
#define ML_CONST 2.9086f
#define SL_CONST 1.898f
#define NTHREADS 1024

// One fused kernel, one workgroup (32 wave32 waves on a single WGP).
// Dynamic LDS layout (floats):
//   sx  [0, 2B+2)        staged copy of x (interleaved u1,u2; +2 pad for pipelining)
//   sol [2B+2, 3B+4)     ol_all (+2 pad)
//   sc  [3B+4, 4B+4)     carry trajectory written by the serial phase
//   rs  [4B+4, +1024)    reduction scratch (sum)
//   rq  [.. , +1024)     reduction scratch (sum of squares)
//   std [.. , +1)        obsstd broadcast slot
//
// Phase 1 (all 1024 threads): copy x -> LDS (float4), compute ol_all -> LDS,
//   tree-reduce y_obs[SPIN:TRAIN] for the ddof=1 std.
// Phase 2 (lane 0 only): the irreducible nonlinear scalar recurrence, reading
//   pipelined ds_load_b64/b32 from LDS, storing only c[t] to LDS.
// Phase 3 (all threads): expand c[t] into all 13 output columns.

__device__ __forceinline__ float sigmoid_core(float oo1, float km, float kb, float c) {
    // sigmoid(b0o + (c-mo)/so*w1o) * oo1  ==  oo1 / (1 + exp(km*c + kb))
    return oo1 / (1.0f + expf(fmaf(km, c, kb)));
}

__global__ __launch_bounds__(NTHREADS) void k_fused(
    const float* __restrict__ x, const int* __restrict__ tl_p,
    const float* __restrict__ y_obs,
    const float* __restrict__ cmean_p, const float* __restrict__ cstd_p,
    const float* __restrict__ wom_p, const float* __restrict__ wlm_p,
    const float* __restrict__ wfm_p,
    const float* __restrict__ b0o_p, const float* __restrict__ w1o_p,
    const float* __restrict__ b0l_p, const float* __restrict__ w2l_p,
    float* __restrict__ out, int B)
{
    extern __shared__ float smem[];
    float* __restrict__ sx   = smem;                 // 2B+2
    float* __restrict__ sol  = sx + (2 * B + 2);     // B+2
    float* __restrict__ sc   = sol + (B + 2);        // B
    float* __restrict__ rs   = sc + B;               // 1024
    float* __restrict__ rq   = rs + NTHREADS;        // 1024
    float* __restrict__ sstd = rq + NTHREADS;        // 1

    const int tid = threadIdx.x;

    // shared scalar params (broadcast loads, L2-cached)
    const float eo = expf(wom_p[0]), el = expf(wlm_p[0]), ef = expf(wfm_p[0]);
    const float denom = eo + el + ef;
    const float oo1 = eo / denom;
    const float ol1 = el / denom;
    const float mo = cmean_p[0], so = cstd_p[0];
    const float w1o = w1o_p[0];
    const float km = -w1o / so;
    const float kb = -b0o_p[0] + mo * (w1o / so);
    const float w2l = w2l_p[0];
    const float kl = w2l / SL_CONST;
    const float bl = b0l_p[0] - (ML_CONST / SL_CONST) * w2l;
    const int tl = tl_p[0];

    // ---------------- Phase 1: staging + reduction -------------------------
    // copy x into LDS with 128-bit transactions
    {
        const float4* __restrict__ xv = reinterpret_cast<const float4*>(x);
        float4* __restrict__ sxv = reinterpret_cast<float4*>(sx);
        const int n4 = (2 * B) >> 2;
        for (int idx = tid; idx < n4; idx += NTHREADS) sxv[idx] = xv[idx];
    }
    // ol_all (reads x straight from global: independent of the LDS copy)
    for (int i = tid; i < B; i += NTHREADS) {
        float u2 = x[2 * i + 1];
        sol[i] = ol1 / (1.0f + expf(-fmaf(u2, kl, bl)));
    }
    // obsstd reduction
    {
        float sum = 0.f, sq = 0.f;
        for (int j = SPIN_LEN + tid; j < TRAIN_LEN; j += NTHREADS) {
            float v = y_obs[j];
            sum += v;
            sq = fmaf(v, v, sq);
        }
        rs[tid] = sum;
        rq[tid] = sq;
    }
    __syncthreads();
    for (int off = NTHREADS / 2; off > 0; off >>= 1) {
        if (tid < off) {
            rs[tid] += rs[tid + off];
            rq[tid] += rq[tid + off];
        }
        __syncthreads();
    }
    if (tid == 0) {
        const float n = (float)(TRAIN_LEN - SPIN_LEN);
        float mean = rs[0] / n;
        float var  = (rq[0] - rs[0] * mean) / (n - 1.0f);
        sstd[0] = sqrtf(fmaxf(var, 0.0f));
    }
    __syncthreads();

    // ---------------- Phase 2: serial recurrence (lane 0) ------------------
    if (tid == 0 && tl < B) {
        const float2* __restrict__ sx2 = reinterpret_cast<const float2*>(sx);
        float c = 0.0f;
        float2 u  = sx2[tl];       // software pipeline: current step's inputs
        float olt = sol[tl];
        for (int t = tl; t < B; ++t) {
            // issue next step's LDS loads before touching the chain
            float2 un  = sx2[t + 1];    // pad slots make t+1==B safe
            float oln  = sol[t + 1];

            sc[t] = c;                                   // pre-update carry

            const float oo  = sigmoid_core(oo1, km, kb, c);
            const float olc = (c > 0.0f) ? fminf(olt, u.y / c) : olt;
            const float f   = (1.0f - oo) - olc;
            c = fmaf(f, c, u.x);

            u = un; olt = oln;
        }
    }
    __syncthreads();

    // ---------------- Phase 3: parallel output expansion -------------------
    const float obsstd = sstd[0];
    for (int i = tid; i < B; i += NTHREADS) {
        if (i < tl) {
            out[0 * B + i] = 0.f;  out[1 * B + i] = 0.f;
            out[2 * B + i] = 0.f;  out[3 * B + i] = 0.f;
            out[4 * B + i] = 0.f;  out[5 * B + i] = 0.f;
            out[6 * B + i] = 0.f;  out[7 * B + i] = 0.f;
            out[8 * B + i] = 0.f;  out[9 * B + i] = 0.f;
            out[10 * B + 2 * i]     = 0.f;
            out[10 * B + 2 * i + 1] = 0.f;
            out[12 * B + i] = 0.f;
        } else {
            const float c   = sc[i];
            const float olt = sol[i];
            const float u2  = sx[2 * i + 1];

            // identical ops to phase 2 => bit-consistent gates
            const float oo  = sigmoid_core(oo1, km, kb, c);
            const float olc = (c > 0.0f) ? fminf(olt, u2 / c) : olt;
            const float f   = (1.0f - oo) - olc;
            const float h   = oo * c;

            out[0 * B + i] = h;          // h_n
            out[1 * B + i] = c;          // c_n
            out[2 * B + i] = olt * c;    // l_n
            out[3 * B + i] = olc * c;    // lc_n
            out[4 * B + i] = 0.f;        // bp_n
            out[5 * B + i] = 0.f;        // gate_ib
            out[6 * B + i] = oo;         // gate_oo
            out[7 * B + i] = olt;        // gate_ol
            out[8 * B + i] = olc;        // gate_olc
            out[9 * B + i] = f;          // gate_f
            out[10 * B + 2 * i]     = h;       // h_nout[:,0]
            out[10 * B + 2 * i + 1] = obsstd;  // h_nout[:,1]
            out[12 * B + i] = obsstd;          // obs_std
        }
    }
}

extern "C" void kernel_launch(void* const* d_in, const int* in_sizes, int n_in,
                              void* d_out, int out_size, void* d_ws, size_t ws_size,
                              hipStream_t stream) {
    (void)n_in; (void)out_size; (void)d_ws; (void)ws_size;
    const float* x      = (const float*)d_in[0];
    /* d_in[1] = epoch (unused) */
    const int*   tl     = (const int*)  d_in[2];
    const float* y_obs  = (const float*)d_in[3];
    const float* cmean  = (const float*)d_in[4];
    const float* cstd   = (const float*)d_in[5];
    const float* wom    = (const float*)d_in[6];
    const float* wlm    = (const float*)d_in[7];
    const float* wfm    = (const float*)d_in[8];
    const float* b0o    = (const float*)d_in[9];
    const float* w1o    = (const float*)d_in[10];
    const float* b0l    = (const float*)d_in[11];
    const float* w2l    = (const float*)d_in[12];
    float* out = (float*)d_out;
    const int B = in_sizes[3];   // y_obs has B elements

    // LDS: sx(2B+2) + sol(B+2) + sc(B) + rs(1024) + rq(1024) + std(1)  [+pad]
    const size_t smem_bytes =
        (size_t)(4 * B + 4 + 2 * NTHREADS + 16) * sizeof(float);   // ~136 KB < 320 KB/WGP

    k_fused<<<1, NTHREADS, smem_bytes, stream>>>(
        x, tl, y_obs, cmean, cstd, wom, wlm, wfm, b0o, w1o, b0l, w2l, out, B);
}